// AffinityBasedAveraging_69587060129919
// MI455X (gfx1250) — compile-verified
//
#include <hip/hip_runtime.h>

#ifndef __has_builtin
#define __has_builtin(x) 0
#endif

#define K_OFF   9
#define N_CH    16
#define IMG_H   512
#define IMG_W   512
#define TILE_H  16
#define TILE_W  32
#define NTHREADS (TILE_H * TILE_W)      // 512 threads = 16 waves (wave32)
#define HALO_H  (TILE_H + 2)
#define HALO_W  (TILE_W + 2)
#define NSPOT   (HALO_H * HALO_W)       // 612 halo pixels per tile
#define CH_STRIDE 20                    // 16 channels padded to 20 dwords (80B: 16B-aligned, low-conflict)
#define NLOAD   (NSPOT * N_CH)          // 9792 floats staged per block

// ---- CDNA5 async global->LDS copy (ASYNCcnt-tracked DMA path) ---------------
__device__ __forceinline__ void async_copy_f32(const float* gsrc, float* ldst) {
#if __has_builtin(__builtin_amdgcn_global_load_async_to_lds_b32)
    __builtin_amdgcn_global_load_async_to_lds_b32(
        (__attribute__((address_space(1))) int*)gsrc,
        (__attribute__((address_space(3))) int*)ldst,
        /*imm offset*/0, /*cpol*/0);
#else
    // Fallback: raw gfx1250 mnemonic. VDST = LDS byte offset (low 32 bits of the
    // flat LDS-aperture address), VADDR = 64-bit global address, SADDR = off.
    unsigned lofs = (unsigned)(size_t)ldst;
    asm volatile("global_load_async_to_lds_b32 %0, %1, off"
                 :: "v"(lofs), "v"(gsrc) : "memory");
#endif
}

__device__ __forceinline__ void wait_async0() {
#if __has_builtin(__builtin_amdgcn_s_wait_asynccnt)
    __builtin_amdgcn_s_wait_asynccnt(0);
#else
    asm volatile("s_wait_asynccnt 0" ::: "memory");
#endif
}
// -----------------------------------------------------------------------------

__global__ __launch_bounds__(NTHREADS)
void aba_stencil_kernel(const float* __restrict__ aff,
                        const float* __restrict__ emb,
                        float* __restrict__ out) {
    __shared__ __align__(16) float tile[NSPOT * CH_STRIDE];  // 47.8 KB

    const int tx  = threadIdx.x;
    const int ty  = threadIdx.y;
    const int tid = ty * TILE_W + tx;
    const int x0  = blockIdx.x * TILE_W;
    const int y0  = blockIdx.y * TILE_H;
    const int b   = blockIdx.z;
    const size_t HW = (size_t)IMG_H * IMG_W;

    const float* embB = emb + (size_t)b * N_CH * HW;

    // ---- Stage (18x34) replicate-clamped halo x 16 channels into LDS -------
    // Linearized so consecutive lanes hit consecutive gx (coalesced global side).
    for (unsigned idx = tid; idx < NLOAD; idx += NTHREADS) {
        unsigned ch = idx / NSPOT;
        unsigned s  = idx - ch * NSPOT;
        unsigned r  = s / HALO_W;
        unsigned cc = s - r * HALO_W;
        int gy = y0 + (int)r - 1;  gy = gy < 0 ? 0 : (gy > IMG_H - 1 ? IMG_H - 1 : gy);
        int gx = x0 + (int)cc - 1; gx = gx < 0 ? 0 : (gx > IMG_W - 1 ? IMG_W - 1 : gx);
        const float* src = embB + (size_t)ch * HW + (size_t)gy * IMG_W + gx;
        async_copy_f32(src, &tile[s * CH_STRIDE + ch]);
    }
    wait_async0();       // drain this wave's ASYNCcnt
    __syncthreads();     // make all waves' LDS writes visible

    const int y = y0 + ty;
    const int x = x0 + tx;

    // ---- Softmax over the 9 offsets + faithful L1 renormalize --------------
    const float* ap = aff + (size_t)b * K_OFF * HW + (size_t)y * IMG_W + x;
    float w[K_OFF];
#pragma unroll
    for (int k = 0; k < K_OFF; ++k) w[k] = ap[k * HW];

    float m = w[0];
#pragma unroll
    for (int k = 1; k < K_OFF; ++k) m = fmaxf(m, w[k]);

    float ssum = 0.0f;
#pragma unroll
    for (int k = 0; k < K_OFF; ++k) { w[k] = __expf(w[k] - m); ssum += w[k]; }

    float inv = 1.0f / ssum;
#pragma unroll
    for (int k = 0; k < K_OFF; ++k) w[k] *= inv;

    float l1 = 0.0f;
#pragma unroll
    for (int k = 0; k < K_OFF; ++k) l1 += fabsf(w[k]);
    float invd = 1.0f / fmaxf(l1, 1e-12f);
#pragma unroll
    for (int k = 0; k < K_OFF; ++k) w[k] *= invd;

    // ---- 9-tap weighted accumulation over 16 channels from LDS -------------
    float acc[N_CH];
#pragma unroll
    for (int c = 0; c < N_CH; ++c) acc[c] = 0.0f;

#pragma unroll
    for (int ky = 0; ky < 3; ++ky) {
#pragma unroll
        for (int kx = 0; kx < 3; ++kx) {
            const float wk = w[ky * 3 + kx];
            const float4* p =
                (const float4*)&tile[((ty + ky) * HALO_W + (tx + kx)) * CH_STRIDE];
            float4 e0 = p[0], e1 = p[1], e2 = p[2], e3 = p[3];
            acc[0]  = fmaf(wk, e0.x, acc[0]);
            acc[1]  = fmaf(wk, e0.y, acc[1]);
            acc[2]  = fmaf(wk, e0.z, acc[2]);
            acc[3]  = fmaf(wk, e0.w, acc[3]);
            acc[4]  = fmaf(wk, e1.x, acc[4]);
            acc[5]  = fmaf(wk, e1.y, acc[5]);
            acc[6]  = fmaf(wk, e1.z, acc[6]);
            acc[7]  = fmaf(wk, e1.w, acc[7]);
            acc[8]  = fmaf(wk, e2.x, acc[8]);
            acc[9]  = fmaf(wk, e2.y, acc[9]);
            acc[10] = fmaf(wk, e2.z, acc[10]);
            acc[11] = fmaf(wk, e2.w, acc[11]);
            acc[12] = fmaf(wk, e3.x, acc[12]);
            acc[13] = fmaf(wk, e3.y, acc[13]);
            acc[14] = fmaf(wk, e3.z, acc[14]);
            acc[15] = fmaf(wk, e3.w, acc[15]);
        }
    }

    // ---- Coalesced stores (consecutive lanes -> consecutive x) -------------
    float* op = out + (size_t)b * N_CH * HW + (size_t)y * IMG_W + x;
#pragma unroll
    for (int c = 0; c < N_CH; ++c) op[c * HW] = acc[c];
}

extern "C" void kernel_launch(void* const* d_in, const int* in_sizes, int n_in,
                              void* d_out, int out_size, void* d_ws, size_t ws_size,
                              hipStream_t stream) {
    const float* aff = (const float*)d_in[0];   // [4, 9, 512, 512] fp32
    const float* emb = (const float*)d_in[1];   // [4, 16, 512, 512] fp32
    float* out = (float*)d_out;                 // [4, 16, 512, 512] fp32

    dim3 grid(IMG_W / TILE_W, IMG_H / TILE_H, 4);   // 16 x 32 x 4 = 2048 blocks
    dim3 block(TILE_W, TILE_H);                     // 512 threads = 16 wave32
    aba_stencil_kernel<<<grid, block, 0, stream>>>(aff, emb, out);
}